// Decoder_83691732730147
// MI455X (gfx1250) — compile-verified
//
#include <hip/hip_runtime.h>
#include <math.h>

#define B_  64
#define NC_ 32
#define NL_ 64
#define E_  256
#define H_  256
#define RS_  (E_ + 4)    // padded LDS row stride for staged context tiles
#define RSR_ (H_ + 4)    // padded LDS row stride for r matrices (bank-conflict-free)
#define CCLIPF 10.0f
#define NTHREADS 256
#define NWAVES 8

typedef __attribute__((ext_vector_type(2))) float v2f;
typedef __attribute__((ext_vector_type(8))) float v8f;

// ---------------- LDS layout (float offsets) ----------------
#define SM_RLOW   0                          // NL_*RSR_
#define SM_RHIGH  (SM_RLOW + NL_*RSR_)       // NC_*RSR_
#define SM_CTX    (SM_RHIGH + NC_*RSR_)      // NL_*RS_   staged node_context cell
#define SM_CC     (SM_CTX + NL_*RS_)         // NC_*RS_   staged cell_context
#define SM_MEAN   (SM_CC + NC_*RS_)          // E_
#define SM_HBARH  (SM_MEAN + E_)             // E_
#define SM_HBARL  (SM_HBARH + E_)            // E_
#define SM_QHI    (SM_HBARL + E_)            // E_
#define SM_QLO    (SM_QHI + E_)              // E_
#define SM_QBUF   (SM_QLO + E_)              // H_
#define SM_INITH  (SM_QBUF + H_)             // E_
#define SM_INITL  (SM_INITH + E_)            // E_
#define SM_RED    (SM_INITL + E_)            // NTHREADS partials
#define SM_U      (SM_RED + NTHREADS)        // NL_ logits
#define SM_SCAL   (SM_U + NL_)               // 16 scalars
#define SM_INTS   (SM_SCAL + 16)             // idx(4) + hmask(32) + lmask(64)
#define SM_TOTAL  (SM_INTS + 112)

// ---------------- deterministic RNG (Gumbel-max sampling) ----------------
__device__ __forceinline__ float hash_u01(unsigned x) {
  x ^= x >> 16; x *= 0x7feb352du; x ^= x >> 15; x *= 0x846ca68bu; x ^= x >> 16;
  return ((float)(x >> 8) + 0.5f) * (1.0f / 16777216.0f);
}
__device__ __forceinline__ float gumbel(unsigned seed, unsigned a, unsigned b,
                                        unsigned c, unsigned d) {
  unsigned x = seed * 0x9E3779B9u ^ (a + 0x85ebca6bu) * 0xc2b2ae35u;
  x = (x ^ (b * 0x27d4eb2fu)) * 0x165667b1u;
  x ^= (c * 0x9E3779B9u) + (d * 0x85ebca6bu) + (x << 6) + (x >> 2);
  float u = hash_u01(x);
  return -logf(-logf(u));
}

// ---- CDNA5 async global->LDS staging: rows of E_ floats, LDS stride RS_ ----
// Coalesced 16B/lane transfers on the ASYNC path (ASYNCcnt), then per-wave
// s_wait_asynccnt; caller must __syncthreads() before reading.
__device__ __forceinline__ void stage_rows_async(const float* __restrict__ g,
                                                 float* __restrict__ lds,
                                                 int rows, int tid) {
  const int chunks = rows * (E_ / 4);          // 16B chunks (uniform multiple of 256)
  for (int c = tid; c < chunks; c += NTHREADS) {
    const int r = c >> 6;                      // E_/4 = 64 chunks per row
    const int f = (c & 63) << 2;               // float offset within row
    unsigned la = (unsigned)(size_t)(const void*)(lds + r * RS_ + f);
    const float* ga = g + (size_t)r * E_ + f;
    asm volatile("global_load_async_to_lds_b128 %0, %1, off"
                 :: "v"(la), "v"(ga) : "memory");
  }
  asm volatile("s_wait_asynccnt 0x0" ::: "memory");
}

// ---------------- WMMA f32 GEMM: out[M x H_] = Alds[M x E_] @ W[E_ x H_] + bias ----
// A staged in LDS (stride RS_, padded -> conflict-free b64 frag reads),
// W row-major global (L2-resident), out in LDS with stride RSR_.
__device__ void gemm_rmat(const float* __restrict__ Alds, int M,
                          const float* __restrict__ W,
                          const float* __restrict__ bias,
                          float* __restrict__ out, int tid)
{
  const int lane = tid & 31;
  const int wave = tid >> 5;
  const int l15  = lane & 15;
  const int lhi  = lane >> 4;
  const int ntiles = (M >> 4) * (H_ >> 4);
  for (int t = wave; t < ntiles; t += NWAVES) {   // uniform per wave: EXEC all-ones
    const int tm = t / (H_ >> 4);
    const int tn = t % (H_ >> 4);
    const int m = (tm << 4) + l15;
    const int n = (tn << 4) + l15;
    const float* arow = Alds + m * RS_;
    const float* wcol = W + n;
    v8f acc = {};
    for (int k0 = 0; k0 < E_; k0 += 4) {
      const int ka = k0 + (lhi << 1);
      v2f av, bvv;
      av.x  = arow[ka];                       // ds_load_b64, conflict-free
      av.y  = arow[ka + 1];
      bvv.x = wcol[(size_t)ka * H_];          // 64B contiguous across half-wave
      bvv.y = wcol[(size_t)(ka + 1) * H_];
      acc = __builtin_amdgcn_wmma_f32_16x16x4_f32(false, av, false, bvv,
                                                  (short)0, acc, false, false);
    }
    const float bn = bias[n];
#pragma unroll
    for (int i = 0; i < 8; ++i)
      out[((tm << 4) + i + (lhi << 3)) * RSR_ + n] = acc[i] + bn;
  }
}

// out[j] = bias[j] + sum_k vec[k] * W[k*H_ + j]   (vec in LDS broadcast, W global)
__device__ __forceinline__ float matvecH(const float* __restrict__ vec,
                                         const float* __restrict__ W,
                                         const float* __restrict__ bias, int j) {
  float acc = bias[j];
  for (int k = 0; k < E_; ++k) acc = fmaf(vec[k], W[(size_t)k * H_ + j], acc);
  return acc;
}

// bias[j] + [a ; b] @ W  where W is (2E_ x E_) row-major; a,b LDS broadcast
__device__ __forceinline__ float matvec_cat(const float* __restrict__ a,
                                            const float* __restrict__ b,
                                            const float* __restrict__ W,
                                            const float* __restrict__ bias, int j) {
  float acc = bias[j];
  for (int k = 0; k < E_; ++k) acc = fmaf(a[k], W[(size_t)k * E_ + j], acc);
  for (int k = 0; k < E_; ++k) acc = fmaf(b[k], W[(size_t)(E_ + k) * E_ + j], acc);
  return acc;
}

// masked pointer scores uv[s] = mask ? -1e9 : CCLIP*tanh( sum_h v[h]*tanh(q[h]+r[s][h]) )
// All NTHREADS threads cooperate: NTHREADS/S partial chunks per target + LDS combine.
template <int S>
__device__ __forceinline__ void score(const float* __restrict__ r,
                                      const float* __restrict__ qb,
                                      const float* __restrict__ vvec,
                                      const int* __restrict__ mask,
                                      float* __restrict__ uvv,
                                      float* __restrict__ red, int tid) {
  constexpr int P  = NTHREADS / S;
  constexpr int HC = H_ / P;
  const int s = tid % S, p = tid / S;          // p uniform per wave
  const float* rr = r + s * RSR_ + p * HC;     // padded stride -> banks rotate
  const float* qq = qb + p * HC;
  const float* vv = vvec + p * HC;
  float acc = 0.f;
  for (int h = 0; h < HC; ++h) acc = fmaf(vv[h], tanhf(qq[h] + rr[h]), acc);
  red[tid] = acc;
  __syncthreads();
  if (tid < S) {
    float u = 0.f;
#pragma unroll
    for (int p2 = 0; p2 < P; ++p2) u += red[tid + p2 * S];
    float lg = CCLIPF * tanhf(u);
    uvv[tid] = (mask[tid] > 0) ? -1.0e9f : lg;
  }
}

__global__ void __launch_bounds__(NTHREADS)
decoder_kernel(const float* __restrict__ node_context,
               const float* __restrict__ cell_context,
               const float* __restrict__ original_data,
               const int*   __restrict__ high_mask,
               const int*   __restrict__ low_mask,
               const float* __restrict__ init_w,
               const float* __restrict__ Wc,  const float* __restrict__ bc,
               const float* __restrict__ Wv,  const float* __restrict__ bv,
               const float* __restrict__ hWq, const float* __restrict__ hbq,
               const float* __restrict__ hWr, const float* __restrict__ hbr,
               const float* __restrict__ hvv,
               const float* __restrict__ low_init_w,
               const float* __restrict__ lWc, const float* __restrict__ lbc,
               const float* __restrict__ lWv, const float* __restrict__ lbv,
               const float* __restrict__ lWq, const float* __restrict__ lbq,
               const float* __restrict__ lWr, const float* __restrict__ lbr,
               const float* __restrict__ lvv,
               float* __restrict__ out)
{
  extern __shared__ float sm[];
  const int tid = threadIdx.x;
  const int b   = blockIdx.x;

  float* rlow   = sm + SM_RLOW;
  float* rhigh  = sm + SM_RHIGH;
  float* ctxLds = sm + SM_CTX;
  float* ccLds  = sm + SM_CC;
  float* meanv  = sm + SM_MEAN;
  float* hbarH  = sm + SM_HBARH;
  float* hbarL  = sm + SM_HBARL;
  float* qHi    = sm + SM_QHI;
  float* qLo    = sm + SM_QLO;
  float* qbuf   = sm + SM_QBUF;
  float* initH  = sm + SM_INITH;
  float* initL  = sm + SM_INITL;
  float* red    = sm + SM_RED;
  float* uv     = sm + SM_U;
  float* scal   = sm + SM_SCAL;  // [0]=logp [1]=rew [2,3]=last_node [4,5]=init_node
  int*   ip     = (int*)(sm + SM_INTS);  // [0]=hi idx [1]=lo idx [2]=cur node
  int*   hmask  = ip + 4;
  int*   lmask  = ip + 4 + NC_;

  const float* cc = cell_context + (size_t)b * NC_ * E_;

  // stage cell_context into LDS on the async path
  stage_rows_async(cc, ccLds, NC_, tid);
  if (tid < 16) scal[tid] = 0.0f;
  if (tid < NC_) hmask[tid] = high_mask[b * NC_ + tid];
  __syncthreads();

  // h_bar (high) = mean(cell_context) @ Wc + bc
  {
    float s0 = 0.f;
    for (int s = 0; s < NC_; ++s) s0 += ccLds[s * RS_ + tid];
    meanv[tid] = s0 * (1.0f / NC_);
  }
  __syncthreads();
  hbarH[tid] = matvecH(meanv, Wc, bc, tid);
  {
    float acc = bv[tid] + hbarH[tid];
    for (int k = 0; k < 2 * E_; ++k) acc = fmaf(init_w[k], Wv[(size_t)k * E_ + tid], acc);
    qHi[tid] = acc;
  }
  // r_high = cell_context @ hWr + hbr (hoisted out of all steps)
  gemm_rmat(ccLds, NC_, hWr, hbr, rhigh, tid);
  __syncthreads();

  for (int step = 0; step < NC_; ++step) {
    // ---- high-level pointer sample ----
    qbuf[tid] = matvecH(qHi, hWq, hbq, tid);
    __syncthreads();
    score<NC_>(rhigh, qbuf, hvv, hmask, uv, red, tid);
    __syncthreads();
    if (tid == 0) {
      float mx = -1.0e30f;
      for (int s = 0; s < NC_; ++s) mx = fmaxf(mx, uv[s]);
      float ssum = 0.f, best = -1.0e30f; int bi = 0;
      for (int s = 0; s < NC_; ++s) {
        ssum += __expf(uv[s] - mx);
        float key = uv[s] + gumbel(42u, (unsigned)step, 0u, (unsigned)s, (unsigned)b);
        if (key > best) { best = key; bi = s; }
      }
      scal[0] += uv[bi] - mx - logf(ssum);
      ip[0] = bi;
      hmask[bi] = 1;
    }
    __syncthreads();
    const int ci = ip[0];
    if (step == 0) initH[tid] = ccLds[ci * RS_ + tid];
    __syncthreads();
    qHi[tid] = hbarH[tid] + matvec_cat(initH, ccLds + ci * RS_, Wv, bv, tid);

    // ---- low-level decode for cell ci ----
    const float* ctx = node_context  + (((size_t)b * NC_ + ci) * NL_) * E_;
    const float* crd = original_data + (((size_t)b * NC_ + ci) * NL_) * 2;
    if (tid < NL_) lmask[tid] = low_mask[((size_t)b * NC_ + ci) * NL_ + tid];
    stage_rows_async(ctx, ctxLds, NL_, tid);   // async ctx tile -> LDS
    __syncthreads();
    // r_low = ctx @ lWr + lbr (hoisted; reference recomputes 64x per cell)
    gemm_rmat(ctxLds, NL_, lWr, lbr, rlow, tid);
    {
      float s0 = 0.f;
      for (int r = 0; r < NL_; ++r) s0 += ctxLds[r * RS_ + tid];
      meanv[tid] = s0 * (1.0f / NL_);
    }
    __syncthreads();
    hbarL[tid] = matvecH(meanv, lWc, lbc, tid);
    {
      float acc = lbv[tid] + hbarL[tid];
      for (int k = 0; k < 2 * E_; ++k)
        acc = fmaf(low_init_w[k], lWv[(size_t)k * E_ + tid], acc);
      qLo[tid] = acc;
    }
    __syncthreads();

    for (int l = 0; l < NL_; ++l) {
      qbuf[tid] = matvecH(qLo, lWq, lbq, tid);
      __syncthreads();
      score<NL_>(rlow, qbuf, lvv, lmask, uv, red, tid);
      __syncthreads();
      if (tid == 0) {
        float mx = -1.0e30f;
        for (int s = 0; s < NL_; ++s) mx = fmaxf(mx, uv[s]);
        float ssum = 0.f, best = -1.0e30f; int bi = 0;
        for (int s = 0; s < NL_; ++s) {
          ssum += __expf(uv[s] - mx);
          float key = uv[s] + gumbel(7u, (unsigned)step, (unsigned)(l + 1),
                                     (unsigned)s, (unsigned)b);
          if (key > best) { best = key; bi = s; }
        }
        scal[0] += uv[bi] - mx - logf(ssum);
        ip[1] = bi;
        lmask[bi] = 1;
        float px, py;
        if (l == 0) { px = crd[0]; py = crd[1]; scal[4] = px; scal[5] = py; }
        else { int cur = ip[2]; px = crd[cur * 2]; py = crd[cur * 2 + 1]; }
        float dx = crd[bi * 2] - px, dy = crd[bi * 2 + 1] - py;
        scal[1] += sqrtf(dx * dx + dy * dy + 1e-12f);
        ip[2] = bi;
      }
      __syncthreads();
      const int ni = ip[1];
      if (l == 0) initL[tid] = ctxLds[ni * RS_ + tid];
      __syncthreads();
      qLo[tid] = hbarL[tid] + matvec_cat(initL, ctxLds + ni * RS_, lWv, lbv, tid);
      __syncthreads();
    }
    // inter-cell reward + last_node update
    if (tid == 0) {
      float dx = scal[4] - scal[2], dy = scal[5] - scal[3];
      scal[1] += sqrtf(dx * dx + dy * dy + 1e-12f);
      int li = ip[2];
      scal[2] = crd[li * 2];
      scal[3] = crd[li * 2 + 1];
    }
    __syncthreads();
  }

  if (tid == 0) {
    out[b]      = scal[0];   // total_log_prob
    out[B_ + b] = scal[1];   // total_reward
  }
}

extern "C" void kernel_launch(void* const* d_in, const int* in_sizes, int n_in,
                              void* d_out, int out_size, void* d_ws, size_t ws_size,
                              hipStream_t stream) {
  (void)in_sizes; (void)n_in; (void)out_size; (void)d_ws; (void)ws_size;
  const float* node_context  = (const float*)d_in[0];
  const float* cell_context  = (const float*)d_in[1];
  const float* original_data = (const float*)d_in[2];
  const int*   high_mask     = (const int*)  d_in[3];
  const int*   low_mask      = (const int*)  d_in[4];
  const float* init_w        = (const float*)d_in[5];
  const float* Wc            = (const float*)d_in[6];
  const float* bc            = (const float*)d_in[7];
  const float* Wv            = (const float*)d_in[8];
  const float* bv            = (const float*)d_in[9];
  const float* hWq           = (const float*)d_in[10];
  const float* hbq           = (const float*)d_in[11];
  const float* hWr           = (const float*)d_in[12];
  const float* hbr           = (const float*)d_in[13];
  const float* hv            = (const float*)d_in[14];
  const float* low_init_w    = (const float*)d_in[15];
  const float* lWc           = (const float*)d_in[16];
  const float* lbc           = (const float*)d_in[17];
  const float* lWv           = (const float*)d_in[18];
  const float* lbv           = (const float*)d_in[19];
  const float* lWq           = (const float*)d_in[20];
  const float* lbq           = (const float*)d_in[21];
  const float* lWr           = (const float*)d_in[22];
  const float* lbr           = (const float*)d_in[23];
  const float* lv            = (const float*)d_in[24];

  const size_t shmem = (size_t)SM_TOTAL * sizeof(float);
  decoder_kernel<<<dim3(B_), dim3(NTHREADS), shmem, stream>>>(
      node_context, cell_context, original_data, high_mask, low_mask,
      init_w, Wc, bc, Wv, bv, hWq, hbq, hWr, hbr, hv,
      low_init_w, lWc, lbc, lWv, lbv, lWq, lbq, lWr, lbr, lv,
      (float*)d_out);
}